// RTRLQuasiLSTMlayer_18038862643944
// MI455X (gfx1250) — compile-verified
//
#include <hip/hip_runtime.h>
#include <math.h>

// Problem sizes (fixed by reference)
#define S_  256
#define B_  32
#define D_  512
#define H_  512
#define BH_ (B_ * H_)      // 16384
#define M_  (S_ * B_)      // 8192 rows for the input projection GEMM

typedef __attribute__((ext_vector_type(2))) float v2f;
typedef __attribute__((ext_vector_type(8))) float v8f;

// ---------------------------------------------------------------------------
// Phase A: pre_z/pre_f[s,b,h] = sum_d x[s,b,d] * wm_{z,f}[h,d]
// M = S*B = 8192, N = H = 512, K = D = 512.  One wave per 16x16 C tile,
// V_WMMA_F32_16X16X4_F32 over 128 K-steps.  A-frag (x) shared by both gates.
// ---------------------------------------------------------------------------
__global__ __launch_bounds__(256) void gemm_pre_kernel(
    const float* __restrict__ x,      // [M_, D_]
    const float* __restrict__ wm_z,   // [H_, D_]
    const float* __restrict__ wm_f,   // [H_, D_]
    float* __restrict__ pre_z,        // [M_, H_]
    float* __restrict__ pre_f)        // [M_, H_]
{
    const int lane  = threadIdx.x & 31;
    const int wid   = blockIdx.x * 8 + (threadIdx.x >> 5);
    const int mtile = wid >> 5;          // 0..511
    const int ntile = wid & 31;          // 0..31
    const int m0 = mtile * 16;
    const int h0 = ntile * 16;
    const int mn   = lane & 15;          // A: row m ; B: col n
    const int half = lane >> 4;          // selects K pair {0,1} vs {2,3}

    const float* xrow  = x    + (size_t)(m0 + mn) * D_;
    const float* wzrow = wm_z + (size_t)(h0 + mn) * D_;
    const float* wfrow = wm_f + (size_t)(h0 + mn) * D_;

    v8f accz = {}; v8f accf = {};
    for (int k0 = 0; k0 < D_; k0 += 4) {
        const int k = k0 + 2 * half;
        v2f a  = *(const v2f*)(xrow  + k);   // A[m][k], A[m][k+1]
        v2f bz = *(const v2f*)(wzrow + k);   // B[k][n], B[k+1][n]
        v2f bf = *(const v2f*)(wfrow + k);
        accz = __builtin_amdgcn_wmma_f32_16x16x4_f32(false, a, false, bz,
                                                     (short)0, accz, false, false);
        accf = __builtin_amdgcn_wmma_f32_16x16x4_f32(false, a, false, bf,
                                                     (short)0, accf, false, false);
    }
#pragma unroll
    for (int r = 0; r < 8; ++r) {
        const int m = m0 + r + 8 * half;     // C layout: M = r + 8*(lane>=16)
        const size_t idx = (size_t)m * H_ + (h0 + mn);
        pre_z[idx] = accz[r];
        pre_f[idx] = accf[r];
    }
}

// ---------------------------------------------------------------------------
// Phase B: scalar RTRL scan over [B,H].  One thread per (b,h).
// Forward: cells + small state outputs (wz/wf/bz/bf, final cell).
// Backward: suffix products of `common`; overwrite pre_z/pre_f in place with
//   g_z[t] = zf_t * prod_{tau>t} common_tau,  g_f[t] = fz_t * (same),
// and emit Ptot = prod_t common_t.
// ---------------------------------------------------------------------------
__global__ __launch_bounds__(256) void rtrl_scan_kernel(
    const float* __restrict__ hidden_prev,  // [B,H]
    const float* __restrict__ wz_state, const float* __restrict__ wf_state,
    const float* __restrict__ bz_state, const float* __restrict__ bf_state,
    const float* __restrict__ wv_z, const float* __restrict__ wv_f,
    const float* __restrict__ bias_z, const float* __restrict__ bias_f,
    float* pre_z, float* pre_f,             // in: projections, out: g_z/g_f
    float* cells,                           // [S,B,H] output
    float* __restrict__ out_newcell,
    float* __restrict__ out_wzn, float* __restrict__ out_wfn,
    float* __restrict__ out_bzn, float* __restrict__ out_bfn,
    float* __restrict__ Ptot)               // [B,H]
{
    const int tid = blockIdx.x * 256 + threadIdx.x;   // = b*H + h
    const int h = tid & (H_ - 1);

    const float wvz = wv_z[h],  wvf = wv_f[h];
    const float biz = bias_z[h], bif = bias_f[h];
    const float c0  = hidden_prev[tid];

    float cell = c0;
    float wz = wz_state[tid], wf = wf_state[tid];
    float bz = bz_state[tid], bf = bf_state[tid];

    for (int t = 0; t < S_; ++t) {
        const size_t idx = (size_t)t * BH_ + tid;
        const float pz = pre_z[idx], pf = pre_f[idx];
        const float z  = tanhf(pz + wvz * cell + biz);
        const float f  = 1.0f / (1.0f + __expf(-(pf + wvf * cell + bif)));
        const float zf = (1.0f - f) * (1.0f - z * z);
        const float fz = (cell - z) * (1.0f - f) * f;
        const float common = f + zf * wvz + fz * wvf;
        wz = cell * zf + common * wz;
        wf = cell * fz + common * wf;
        bz = zf + common * bz;
        bf = fz + common * bf;
        cell = cell * f + (1.0f - f) * z;
        cells[idx] = cell;
    }
    out_newcell[tid] = cell;
    out_wzn[tid] = wz;  out_wfn[tid] = wf;
    out_bzn[tid] = bz;  out_bfn[tid] = bf;

    // Backward: recompute gates from pre + cells (this thread's own writes).
    float suffix = 1.0f;
    for (int t = S_ - 1; t >= 0; --t) {
        const size_t idx = (size_t)t * BH_ + tid;
        const float cprev = (t == 0) ? c0 : cells[idx - BH_];
        const float pz = pre_z[idx], pf = pre_f[idx];
        const float z  = tanhf(pz + wvz * cprev + biz);
        const float f  = 1.0f / (1.0f + __expf(-(pf + wvf * cprev + bif)));
        const float zf = (1.0f - f) * (1.0f - z * z);
        const float fz = (cprev - z) * (1.0f - f) * f;
        const float common = f + zf * wvz + fz * wvf;
        pre_z[idx] = zf * suffix;       // g_z[t]
        pre_f[idx] = fz * suffix;       // g_f[t]
        suffix *= common;
    }
    Ptot[tid] = suffix;
}

// ---------------------------------------------------------------------------
// Phase C: Zn[b,h,d] = Ptot[b,h]*Z0[b,h,d] + sum_t g_z[t,b,h]*x[t,b,d]
// (same for Fn with g_f).  Per-b GEMM (H x S)*(S x D) via wmma f32 16x16x4.
// B-frag (x) shared by both accumulators.  One wave per 16x16 output tile.
// ---------------------------------------------------------------------------
__global__ __launch_bounds__(256) void rtrl_outer_kernel(
    const float* __restrict__ x,     // [S,B,D]
    const float* __restrict__ gz,    // [S,B,H] (pre_z buffer, now g_z)
    const float* __restrict__ gf,    // [S,B,H]
    const float* __restrict__ Z0,    // [B,H,D]
    const float* __restrict__ F0,    // [B,H,D]
    const float* __restrict__ Ptot,  // [B,H]
    float* __restrict__ Zn, float* __restrict__ Fn)
{
    const int lane = threadIdx.x & 31;
    const int wid  = blockIdx.x * 8 + (threadIdx.x >> 5);
    const int b  = wid >> 10;            // 0..31
    const int ht = (wid >> 5) & 31;      // 0..31
    const int dt = wid & 31;             // 0..31
    const int h0 = ht * 16, d0 = dt * 16;
    const int mn   = lane & 15;          // A: h-row ; B: d-col
    const int half = lane >> 4;

    v8f accz = {}; v8f accf = {};
    for (int s0 = 0; s0 < S_; s0 += 4) {
        const int s = s0 + 2 * half;
        const size_t xb0 = ((size_t)s * B_ + b) * D_ + d0 + mn;
        const size_t gb0 = ((size_t)s * B_ + b) * H_ + h0 + mn;
        v2f bx, az, af;
        bx[0] = x[xb0];            bx[1] = x[xb0 + (size_t)B_ * D_];
        az[0] = gz[gb0];           az[1] = gz[gb0 + (size_t)B_ * H_];
        af[0] = gf[gb0];           af[1] = gf[gb0 + (size_t)B_ * H_];
        accz = __builtin_amdgcn_wmma_f32_16x16x4_f32(false, az, false, bx,
                                                     (short)0, accz, false, false);
        accf = __builtin_amdgcn_wmma_f32_16x16x4_f32(false, af, false, bx,
                                                     (short)0, accf, false, false);
    }
#pragma unroll
    for (int r = 0; r < 8; ++r) {
        const int h = h0 + r + 8 * half;
        const size_t idx = ((size_t)b * H_ + h) * D_ + (d0 + mn);
        const float p = Ptot[b * H_ + h];
        Zn[idx] = p * Z0[idx] + accz[r];
        Fn[idx] = p * F0[idx] + accf[r];
    }
}

// ---------------------------------------------------------------------------
// Host launcher
// ---------------------------------------------------------------------------
extern "C" void kernel_launch(void* const* d_in, const int* in_sizes, int n_in,
                              void* d_out, int out_size, void* d_ws, size_t ws_size,
                              hipStream_t stream) {
    const float* x           = (const float*)d_in[0];
    const float* hidden_prev = (const float*)d_in[1];
    const float* Z_state     = (const float*)d_in[2];
    const float* F_state     = (const float*)d_in[3];
    const float* wz_state    = (const float*)d_in[4];
    const float* wf_state    = (const float*)d_in[5];
    const float* bz_state    = (const float*)d_in[6];
    const float* bf_state    = (const float*)d_in[7];
    const float* wm_z        = (const float*)d_in[8];
    const float* wm_f        = (const float*)d_in[9];
    const float* wv_z        = (const float*)d_in[10];
    const float* wv_f        = (const float*)d_in[11];
    const float* bias_z      = (const float*)d_in[12];
    const float* bias_f      = (const float*)d_in[13];

    float* out = (float*)d_out;
    // Output tuple, flat in return order
    float* cells = out;                                  // S*B*H = 4194304
    float* newc  = out + (size_t)S_ * BH_;               // B*H
    float* Zn    = newc + BH_;                           // B*H*D
    float* Fn    = Zn + (size_t)BH_ * D_;                // B*H*D
    float* wzn   = Fn + (size_t)BH_ * D_;
    float* wfn   = wzn + BH_;
    float* bzn   = wfn + BH_;
    float* bfn   = bzn + BH_;

    // Workspace: pre_z / pre_f (reused in place as g_z / g_f) + Ptot (~33.6 MB)
    float* ws    = (float*)d_ws;
    float* pre_z = ws;
    float* pre_f = ws + (size_t)S_ * BH_;
    float* Ptot  = pre_f + (size_t)S_ * BH_;

    // Phase A: input projections (512 m-tiles x 32 h-tiles = 16384 waves)
    gemm_pre_kernel<<<2048, 256, 0, stream>>>(x, wm_z, wm_f, pre_z, pre_f);

    // Phase B: scalar scan over B*H = 16384 threads
    rtrl_scan_kernel<<<64, 256, 0, stream>>>(
        hidden_prev, wz_state, wf_state, bz_state, bf_state,
        wv_z, wv_f, bias_z, bias_f,
        pre_z, pre_f, cells, newc, wzn, wfn, bzn, bfn, Ptot);

    // Phase C: collapsed sensitivity update (32 b x 32 h-tiles x 32 d-tiles)
    rtrl_outer_kernel<<<4096, 256, 0, stream>>>(
        x, pre_z, pre_f, Z_state, F_state, Ptot, Zn, Fn);
}